// WAUC_14534169330300
// MI455X (gfx1250) — compile-verified
//
#include <hip/hip_runtime.h>

// ---------------------------------------------------------------------------
// WAUC reduction for MI455X (gfx1250, wave32).
// Memory-bound: 147.5 MB single-pass stream -> ~6.3 us at 23.3 TB/s.
// - 128-bit non-temporal loads (clang ext_vector float4), 5-deep grid-stride.
// - suf[] LUT replaced by closed form: suf(i) = ((101-i)(102-i)/2 - 1)/100.
// - Cross-lane reduction via V_WMMA_F32_16X16X4_F32 (full f32 precision),
//   deterministic two-stage reduction (no float atomics).
// ---------------------------------------------------------------------------

typedef float v2f __attribute__((ext_vector_type(2)));
typedef float v4f __attribute__((ext_vector_type(4)));
typedef float v8f __attribute__((ext_vector_type(8)));

#define HW        (720 * 1280)        // 921600 pixels per image
#define NBATCH    8
#define NGROUPS   ((NBATCH * HW) / 4) // 1,843,200 float4 groups
#define TPB       256
#define NBLOCKS   1440
#define GSTRIDE   (NBLOCKS * TPB)     // 368,640 ; NGROUPS / GSTRIDE == 5 exactly
#define NITER     5
#define WAVES_PER_BLOCK (TPB / 32)

// Full 32-lane f32 sum using the f32 WMMA.
// A (16x4): lane l<16 holds A[l][0..1], lane l+16 holds A[l][2..3] -> put v in
// both components so rowsum[m] = 2*(v[m] + v[m+16]). B = ones (layout-proof).
// D[m][n] = rowsum[m]; c[r] holds M=r (lanes<16) / M=r+8 (lanes>=16).
// sum_r c[r] then fold halves with shfl_xor(16): every lane gets 2*total.
__device__ __forceinline__ float wave_allreduce_f32(float v) {
    v2f a; a[0] = v;    a[1] = v;
    v2f b; b[0] = 1.0f; b[1] = 1.0f;
    v8f c = {};
    c = __builtin_amdgcn_wmma_f32_16x16x4_f32(
            /*neg_a=*/false, a, /*neg_b=*/false, b,
            /*c_mod=*/(short)0, c, /*reuse_a=*/false, /*reuse_b=*/false);
    float s = ((c[0] + c[1]) + (c[2] + c[3])) + ((c[4] + c[5]) + (c[6] + c[7]));
    s += __shfl_xor(s, 16, 32);
    return s * 0.5f;
}

// suf[clip(ceil(20*dist),0,100)] in closed form (exact at the endpoints:
// c=0 -> 51.5, c=100 -> 0).
__device__ __forceinline__ float suf_of_dist(float dx, float dy) {
    float d = sqrtf(dx * dx + dy * dy);
    float c = ceilf(d * 20.0f);           // d >= 0, so c >= 0 already
    c = fminf(c, 100.0f);
    return ((101.0f - c) * (102.0f - c) * 0.5f - 1.0f) * 0.01f;
}

__global__ __launch_bounds__(TPB) void wauc_partial(
        const float* __restrict__ in_flow,   // (8, 2, 720, 1280)
        const float* __restrict__ tg_flow,   // (8, 3, 720, 1280)
        float* __restrict__ ws)              // [0..NBLOCKS): err, [NBLOCKS..2N): mask
{
    const int tid0 = blockIdx.x * TPB + threadIdx.x;
    float err_acc  = 0.0f;
    float mask_acc = 0.0f;

#pragma unroll
    for (int it = 0; it < NITER; ++it) {
        const int g = tid0 + it * GSTRIDE;   // float4 group index
        const int p = g * 4;                 // global pixel index
        const int bb = p / HW;               // batch
        const int q  = p - bb * HW;          // pixel within image (multiple of 4)

        const float* ib = in_flow + 2 * bb * HW;
        const float* tb = tg_flow + 3 * bb * HW;

        const v4f ix = __builtin_nontemporal_load((const v4f*)(ib + q));
        const v4f iy = __builtin_nontemporal_load((const v4f*)(ib + HW + q));
        const v4f tx = __builtin_nontemporal_load((const v4f*)(tb + q));
        const v4f ty = __builtin_nontemporal_load((const v4f*)(tb + HW + q));
        const v4f mk = __builtin_nontemporal_load((const v4f*)(tb + 2 * HW + q));

        err_acc += suf_of_dist(tx.x - ix.x, ty.x - iy.x);
        err_acc += suf_of_dist(tx.y - ix.y, ty.y - iy.y);
        err_acc += suf_of_dist(tx.z - ix.z, ty.z - iy.z);
        err_acc += suf_of_dist(tx.w - ix.w, ty.w - iy.w);
        mask_acc += (mk.x + mk.y) + (mk.z + mk.w);
    }

    // EXEC is all-1s here (no divergence above) -> WMMA legal.
    const float werr  = wave_allreduce_f32(err_acc);
    const float wmask = wave_allreduce_f32(mask_acc);

    __shared__ float lerr[WAVES_PER_BLOCK];
    __shared__ float lmask[WAVES_PER_BLOCK];
    const int lane = threadIdx.x & 31;
    const int wave = threadIdx.x >> 5;
    if (lane == 0) { lerr[wave] = werr; lmask[wave] = wmask; }
    __syncthreads();
    if (threadIdx.x == 0) {
        float e = 0.0f, m = 0.0f;
#pragma unroll
        for (int i = 0; i < WAVES_PER_BLOCK; ++i) { e += lerr[i]; m += lmask[i]; }
        ws[blockIdx.x]           = e;
        ws[NBLOCKS + blockIdx.x] = m;
    }
}

__global__ __launch_bounds__(TPB) void wauc_final(
        const float* __restrict__ ws, float* __restrict__ out)
{
    float e = 0.0f, m = 0.0f;
    for (int i = threadIdx.x; i < NBLOCKS; i += TPB) {   // 6 trips for threads < 160, 5 for the rest
        e += ws[i];
        m += ws[NBLOCKS + i];
    }
    const float werr  = wave_allreduce_f32(e);
    const float wmask = wave_allreduce_f32(m);

    __shared__ float lerr[WAVES_PER_BLOCK];
    __shared__ float lmask[WAVES_PER_BLOCK];
    const int lane = threadIdx.x & 31;
    const int wave = threadIdx.x >> 5;
    if (lane == 0) { lerr[wave] = werr; lmask[wave] = wmask; }
    __syncthreads();
    if (threadIdx.x == 0) {
        float te = 0.0f, tm = 0.0f;
#pragma unroll
        for (int i = 0; i < WAVES_PER_BLOCK; ++i) { te += lerr[i]; tm += lmask[i]; }
        // n_weights = sum((101-i)/100, i=0..99) = 51.5
        out[0] = 100.0f * te / (tm * 51.5f);
    }
}

extern "C" void kernel_launch(void* const* d_in, const int* in_sizes, int n_in,
                              void* d_out, int out_size, void* d_ws, size_t ws_size,
                              hipStream_t stream) {
    const float* in_flow = (const float*)d_in[0];   // (8,2,720,1280) f32
    const float* tg_flow = (const float*)d_in[1];   // (8,3,720,1280) f32
    float* out = (float*)d_out;                     // scalar f32
    float* ws  = (float*)d_ws;                      // 2*NBLOCKS floats used

    wauc_partial<<<NBLOCKS, TPB, 0, stream>>>(in_flow, tg_flow, ws);
    wauc_final<<<1, TPB, 0, stream>>>(ws, out);
}